// RWKV_Tmix_x070_71932112273785
// MI455X (gfx1250) — compile-verified
//
#include <hip/hip_runtime.h>

// RWKV-7 chunked WKV forward for MI455X (gfx1250), fp32 WMMA path with
// async global->LDS pipelining.
//
// Roofline: 940 MB @ 23.3 TB/s => ~40us floor; 26 GFLOP fp32 matmul is
// negligible => stay fp32 (V_WMMA_F32_16X16X4_F32) for exact reference
// numerics. 128 blocks = one (b,h) sequential scan each (~1 per WGP), so the
// per-chunk serial chain is the limiter: next chunk's 24KB of inputs are
// fetched with GLOBAL_LOAD_ASYNC_TO_LDS_B128 (ASYNCcnt) while WMMA phases of
// the current chunk run. State update folds fw: S_new = (S + sv^T@kwi +
// u^T@bwi) * fw[col], so transposed operands are stored unscaled and reused
// by Phase B, eliminating the row-major kwi/bwi copies.

typedef __attribute__((ext_vector_type(2))) float v2f;
typedef __attribute__((ext_vector_type(8))) float v8f;

#define T_LEN 4096
#define HC_DIM 2048
#define NHEAD 32
#define CH 64          // head size
#define DT 16          // chunk length
#define NT (T_LEN / DT)

#define SS 65          // stride of [16][64]-ish row-major arrays and S (banks: 65n%64=n)
#define TS 17          // stride of transposed [64][16] arrays and 16x16 mats

// D = A(16x4) * B(4x16) + C, fp32.
// Layouts (ISA 7.12.2): A: lane&15 = row M, VGPRs hold K = 2*(lane>>4)+{0,1}.
// B assumed mirror: lane&15 = col N. C/D: VGPR r -> M = r + 8*(lane>>4), N = lane&15.
__device__ __forceinline__ v8f wmma4(v2f a, v2f b, v8f c) {
  return __builtin_amdgcn_wmma_f32_16x16x4_f32(
      false, a, false, b, (short)0, c, false, false);
}

// Fragment with K contiguous in the row: lane reads 2 floats at
// row = rbase + (lane&15), col = cbase + 2*(lane>>4).  -> ds_load_2addr_b64
__device__ __forceinline__ v2f frag2(const float* p, int rs, int rbase,
                                     int cbase, int lane) {
  const float* q = p + (rbase + (lane & 15)) * rs + cbase + ((lane >> 4) << 1);
  return v2f{q[0], q[1]};
}

// Fragment from a transposed [c][t] array used as B with K=c: lane n needs
// elements (t=n, c=cbase+2*(lane>>4)+{0,1}) -> two b32 reads, stride rs apart.
__device__ __forceinline__ v2f frag2T(const float* p, int rs, int cbase,
                                      int lane) {
  const float* q = p + (cbase + ((lane >> 4) << 1)) * rs + (lane & 15);
  return v2f{q[0], q[rs]};
}

// Low 32 bits of a generic pointer to LDS == LDS byte offset (aperture holds
// the high bits), which is what async-to-LDS wants in its VDST VGPR.
__device__ __forceinline__ unsigned lds_u32(const void* p) {
  return (unsigned)(unsigned long long)p;
}

// GVS-mode async copy: copies 16B per lane from global address sbase plus offset voff to LDS offset lds_off,
// tracked by ASYNCcnt.
__device__ __forceinline__ void async_load_b128(unsigned lds_off,
                                                const void* sbase,
                                                unsigned voff) {
  asm volatile("global_load_async_to_lds_b128 %0, %1, %2"
               :: "v"(lds_off), "v"(voff), "s"(sbase)
               : "memory");
}

__device__ __forceinline__ void wait_async0() {
  asm volatile("s_wait_asynccnt 0x0" ::: "memory");
}

__global__ __launch_bounds__(128, 1) void wkv7_fwd(
    const float* __restrict__ q, const float* __restrict__ w,
    const float* __restrict__ k, const float* __restrict__ v,
    const float* __restrict__ a, const float* __restrict__ b,
    float* __restrict__ y) {
  extern __shared__ float lds[];
  float* S     = lds;                 // [64][SS] state (v,k)
  float* wqL   = S    + CH * SS;      // [16][SS]
  float* waL   = wqL  + DT * SS;      // [16][SS]
  float* svT   = waL  + DT * SS;      // [64][TS]  v^T   (K=t)
  float* uT    = svT  + CH * TS;      // [64][TS]  u^T   (K=t)
  float* kwiT  = uT   + CH * TS;      // [64][TS]  kwi^T (unscaled)
  float* bwiT  = kwiT + CH * TS;      // [64][TS]  bwi^T (unscaled)
  float* abuT  = bwiT + CH * TS;      // [64][TS]  ab_u^T
  float* abL   = abuT + CH * TS;      // [16][TS]
  float* akL   = abL  + DT * TS;
  float* qkL   = akL  + DT * TS;
  float* qbL   = qkL  + DT * TS;
  float* MinvL = qbL  + DT * TS;
  float* fwL   = MinvL + DT * TS;     // [64]
  float* raw   = fwL + CH;            // [6][16*64] async-staged w,q,k,v,a,b

  const int tid = threadIdx.x;
  const int wave = tid >> 5;
  const int lane = tid & 31;
  const int nlo = lane & 15;
  const int mhi = (lane >> 4) << 3;
  const int vb = wave * 16;

  const int bi = blockIdx.x / NHEAD;
  const int hi = blockIdx.x % NHEAD;
  const size_t gbase = (size_t)bi * T_LEN * HC_DIM + (size_t)hi * CH;

  // Per-thread async-copy geometry: unit u (16B) -> row t=u>>4, colgrp=u&15.
  const unsigned voff0 = ((unsigned)(tid >> 4) * (HC_DIM * 4)) + ((tid & 15) << 4);
  const unsigned voff1 = voff0 + 8u * (HC_DIM * 4);  // u+128 => t+8

#define ISSUE_CHUNK(GELEM)                                                   \
  {                                                                          \
    const float* bases[6] = {w, q, k, v, a, b};                              \
    _Pragma("unroll")                                                        \
    for (int arr = 0; arr < 6; ++arr) {                                      \
      unsigned l0 = lds_u32(raw + arr * (DT * CH)) + (unsigned)tid * 16u;    \
      async_load_b128(l0, (const void*)(bases[arr] + (GELEM)), voff0);       \
      async_load_b128(l0 + 128u * 16u, (const void*)(bases[arr] + (GELEM)),  \
                      voff1);                                                \
    }                                                                        \
  }

  for (int i = tid; i < CH * SS; i += 128) S[i] = 0.f;
  ISSUE_CHUNK(gbase)  // prefetch chunk 0

  for (int nt = 0; nt < NT; ++nt) {
    const int t0 = nt * DT;

    wait_async0();      // this wave's raw-chunk copies done
    __syncthreads();    // ...and everyone else's

    // ---- Phase A: decay scan + operand staging (all reads from LDS raw) ---
    if (tid < CH) {
      const int c = tid;
      const float* rw = raw + 0 * (DT * CH);
      const float* rq = raw + 1 * (DT * CH);
      const float* rk = raw + 2 * (DT * CH);
      const float* ra = raw + 4 * (DT * CH);
      const float* rb = raw + 5 * (DT * CH);
      float incl = 1.f;
      #pragma unroll
      for (int t = 0; t < DT; ++t) {
        const float dec = __expf(-__expf(rw[t * CH + c]));
        const float prev = incl;          // non-inclusive prefix
        incl *= dec;                      // inclusive prefix
        const float rinv = 1.f / incl;
        wqL[t * SS + c] = rq[t * CH + c] * incl;
        waL[t * SS + c] = ra[t * CH + c] * prev;
        kwiT[c * TS + t] = rk[t * CH + c] * rinv;   // unscaled
        bwiT[c * TS + t] = rb[t * CH + c] * rinv;   // unscaled
      }
      fwL[c] = incl;                      // whole-chunk decay
    } else {
      const int c = tid - CH;
      const float* rv = raw + 3 * (DT * CH);
      #pragma unroll
      for (int t = 0; t < DT; ++t) svT[c * TS + t] = rv[t * CH + c];
    }
    __syncthreads();    // staged arrays ready; raw fully consumed

    if (nt + 1 < NT) ISSUE_CHUNK(gbase + (size_t)(t0 + DT) * HC_DIM)

    // ---- Phase B: ab/ak/qk/qb (one 16x16 product per wave), masked --------
    {
      const float* Am = (wave < 2) ? waL : wqL;
      const float* Bt = (wave == 0 || wave == 3) ? bwiT : kwiT;
      v8f acc = {};
      #pragma unroll
      for (int k0 = 0; k0 < CH; k0 += 4)
        acc = wmma4(frag2(Am, SS, 0, k0, lane), frag2T(Bt, TS, k0, lane), acc);
      float* dst = (wave == 0) ? abL : (wave == 1) ? akL
                 : (wave == 2) ? qkL : qbL;
      const bool strict = (wave < 2);
      #pragma unroll
      for (int r = 0; r < 8; ++r) {
        const int m = r + mhi;
        const bool keep = strict ? (m > nlo) : (m >= nlo);
        dst[m * TS + nlo] = keep ? acc[r] : 0.f;
      }
    }
    __syncthreads();

    // ---- Phase C: ab_u = ak@sv + wa@S^T (per-wave col block); Minv --------
    {
      v8f acc = {};
      #pragma unroll
      for (int k0 = 0; k0 < DT; k0 += 4)
        acc = wmma4(frag2(akL, TS, 0, k0, lane),
                    frag2(svT, TS, vb, k0, lane), acc);
      #pragma unroll
      for (int k0 = 0; k0 < CH; k0 += 4)
        acc = wmma4(frag2(waL, SS, 0, k0, lane),
                    frag2(S, SS, vb, k0, lane), acc);
      #pragma unroll
      for (int r = 0; r < 8; ++r)
        abuT[(vb + nlo) * TS + (r + mhi)] = acc[r];
      if (wave == 0 && lane < DT) {       // Minv = (I-ab)^-1 (unit lower)
        const int c = lane;
        MinvL[c] = (c == 0) ? 1.f : 0.f;
        for (int t = 1; t < DT; ++t) {
          float s = (t == c) ? 1.f : 0.f;
          for (int j = 0; j < t; ++j) s += abL[t * TS + j] * MinvL[j * TS + c];
          MinvL[t * TS + c] = s;
        }
      }
    }
    __syncthreads();

    // ---- Phase D: u = Minv@ab_u ; y partials not needing u ----------------
    v8f yacc = {};
    {
      v8f uacc = {};
      #pragma unroll
      for (int k0 = 0; k0 < DT; k0 += 4)
        uacc = wmma4(frag2(MinvL, TS, 0, k0, lane),
                     frag2(abuT, TS, vb, k0, lane), uacc);
      #pragma unroll
      for (int r = 0; r < 8; ++r)
        uT[(vb + nlo) * TS + (r + mhi)] = uacc[r];
      #pragma unroll
      for (int k0 = 0; k0 < DT; k0 += 4)
        yacc = wmma4(frag2(qkL, TS, 0, k0, lane),
                     frag2(svT, TS, vb, k0, lane), yacc);
      #pragma unroll
      for (int k0 = 0; k0 < CH; k0 += 4)
        yacc = wmma4(frag2(wqL, SS, 0, k0, lane),
                     frag2(S, SS, vb, k0, lane), yacc);
    }
    // uT / S rows [vb,vb+16) are wave-private: no barrier needed before E.

    // ---- Phase E: y += qb@u, store y; S = (S + sv^T@kwi + u^T@bwi)*fw -----
    {
      #pragma unroll
      for (int k0 = 0; k0 < DT; k0 += 4)
        yacc = wmma4(frag2(qbL, TS, 0, k0, lane),
                     frag2(uT, TS, vb, k0, lane), yacc);
      #pragma unroll
      for (int r = 0; r < 8; ++r) {
        const int m = r + mhi;
        y[gbase + (size_t)(t0 + m) * HC_DIM + vb + nlo] = yacc[r];
      }
      #pragma unroll
      for (int kb = 0; kb < CH; kb += 16) {
        v8f sacc = {};
        #pragma unroll
        for (int k0 = 0; k0 < DT; k0 += 4) {
          sacc = wmma4(frag2(svT, TS, vb, k0, lane),
                       frag2(kwiT, TS, kb, k0, lane), sacc);
          sacc = wmma4(frag2(uT, TS, vb, k0, lane),
                       frag2(bwiT, TS, kb, k0, lane), sacc);
        }
        #pragma unroll
        for (int r = 0; r < 8; ++r) {
          const int row = vb + r + mhi;
          const int col = kb + nlo;
          S[row * SS + col] = (S[row * SS + col] + sacc[r]) * fwL[col];
        }
      }
    }
    __syncthreads();  // protect per-chunk LDS arrays before next Phase A
  }
#undef ISSUE_CHUNK
}

extern "C" void kernel_launch(void* const* d_in, const int* in_sizes, int n_in,
                              void* d_out, int out_size, void* d_ws,
                              size_t ws_size, hipStream_t stream) {
  // setup_inputs order: r, w, k, v, a, b ; reference maps q := r.
  const float* r = (const float*)d_in[0];
  const float* w = (const float*)d_in[1];
  const float* k = (const float*)d_in[2];
  const float* v = (const float*)d_in[3];
  const float* a = (const float*)d_in[4];
  const float* b = (const float*)d_in[5];
  float* y = (float*)d_out;

  const long total = (long)in_sizes[0];
  const int B = (int)(total / ((long)T_LEN * HC_DIM));

  // Dynamic LDS: 19248 floats = 76,992 B (fine vs CDNA5's 320 KB/WGP).
  const size_t lds_bytes =
      (size_t)(CH * SS + 2 * DT * SS + 5 * CH * TS + 5 * DT * TS + CH +
               6 * DT * CH) * sizeof(float);

  dim3 grid(B * NHEAD);   // one block per (batch, head) sequential scan
  dim3 block(128);        // 4 wave32 waves; wave j owns value cols [16j,16j+16)
  wkv7_fwd<<<grid, block, lds_bytes, stream>>>(r, w, k, v, a, b, y);
}